// SPR_RGCN_TokenBlind_88648124990706
// MI455X (gfx1250) — compile-verified
//
#include <hip/hip_runtime.h>

typedef __attribute__((ext_vector_type(2))) float v2f;
typedef __attribute__((ext_vector_type(4))) float v4f;
typedef __attribute__((ext_vector_type(8))) float v8f;

#define NDIM 64
#define NRELS 3

// ---------------- embedding lookup: h[n][d] = embed_w[x[n]][d] ----------------
__global__ void embed_kernel(const int* __restrict__ x, const float* __restrict__ ew,
                             float* __restrict__ h, int n64) {
  int i = blockIdx.x * blockDim.x + threadIdx.x;
  if (i < n64) {
    int n = i >> 6, d = i & 63;
    h[i] = ew[x[n] * NDIM + d];
  }
}

__global__ void zero_u32_kernel(unsigned* __restrict__ p, int n) {
  int i = blockIdx.x * blockDim.x + threadIdx.x;
  if (i < n) p[i] = 0u;
}

__global__ void zero_f32_kernel(float* __restrict__ p, int n) {
  int i = blockIdx.x * blockDim.x + threadIdx.x;
  if (i < n) p[i] = 0.0f;
}

// ---------------- per-(dst, relation) in-degree counts ----------------
__global__ void count_kernel(const int* __restrict__ dst, const int* __restrict__ et,
                             unsigned* __restrict__ cnt, int E) {
  int e = blockIdx.x * blockDim.x + threadIdx.x;
  if (e < E) atomicAdd(&cnt[(long)dst[e] * NRELS + et[e]], 1u);
}

// ---------------- WMMA GEMM: tr[r] = h@W[r]; acc = h@root + b ----------------
// Weights (3xW + root = 64KB) staged once per block into LDS, pre-paired as
// v2f{W[k][n], W[k+1][n]} B-fragments -> each WMMA K-step is one ds_load_b64.
// One wave per 16-node row tile; 4 matrices x 4 col-tiles x 16 K-steps of
// v_wmma_f32_16x16x4_f32 per wave.
__global__ void __launch_bounds__(256) rgcn_gemm_kernel(
    const float* __restrict__ hin,
    const float* __restrict__ W,    // [3][64][64]
    const float* __restrict__ root, // [64][64]
    const float* __restrict__ bias, // [64]
    float* __restrict__ tr,         // [3][N][64]
    float* __restrict__ acc,        // [N][64]
    int N, int ntiles) {
  __shared__ v2f lds_b[4 * 32 * NDIM];   // [m][kb2][n], 64 KB

  // ---- cooperative stage of all 4 weight matrices ----
  for (int idx = threadIdx.x; idx < 4 * 32 * NDIM; idx += 256) {
    int m   = idx >> 11;        // 0..3
    int kb2 = (idx >> 6) & 31;  // even-k pair index
    int n   = idx & 63;
    const float* Wm = (m < 3) ? (W + m * NDIM * NDIM) : root;
    v2f v;
    v.x = Wm[(2 * kb2) * NDIM + n];
    v.y = Wm[(2 * kb2 + 1) * NDIM + n];
    lds_b[idx] = v;
  }
  __syncthreads();

  const int wave = (int)((blockIdx.x * blockDim.x + threadIdx.x) >> 5);
  const int lane = threadIdx.x & 31;

  if (wave < ntiles) {                    // wave-uniform: EXEC all-1s inside
    const int r0    = wave * 16;
    const int mrow  = lane & 15;
    const int khalf = lane >> 4;          // 0: K pair {k,k+1}; 1: {k+2,k+3}
    const bool full = (r0 + 16 <= N);

    int arow = r0 + mrow;
    if (arow >= N) arow = N - 1;          // clamp loads only
    const v2f* hrow2 = (const v2f*)(hin + (long)arow * NDIM);

    v2f a[16];
#pragma unroll
    for (int j = 0; j < 16; ++j) a[j] = hrow2[2 * j + khalf];

    for (int m = 0; m < 4; ++m) {
      float* outp = (m < 3) ? (tr + (long)m * N * NDIM) : acc;
      const v2f* bbase = lds_b + (m << 5) * NDIM;
      for (int nt = 0; nt < 4; ++nt) {
        const int ncol = nt * 16 + mrow;
        float cinit = (m == 3) ? bias[ncol] : 0.0f;
        v8f c;
#pragma unroll
        for (int i = 0; i < 8; ++i) c[i] = cinit;
#pragma unroll
        for (int j = 0; j < 16; ++j) {
          v2f b = bbase[(2 * j + khalf) * NDIM + ncol];   // ds_load_b64
          c = __builtin_amdgcn_wmma_f32_16x16x4_f32(
                  /*neg_a=*/false, a[j], /*neg_b=*/false, b,
                  /*c_mod=*/(short)0, c, /*reuse_a=*/false, /*reuse_b=*/false);
        }
        if (full) {
#pragma unroll
          for (int i = 0; i < 8; ++i)
            outp[(long)(r0 + i + khalf * 8) * NDIM + ncol] = c[i];
        } else {
#pragma unroll
          for (int i = 0; i < 8; ++i) {
            int row = r0 + i + khalf * 8;
            if (row < N) outp[(long)row * NDIM + ncol] = c[i];
          }
        }
      }
    }
  }
}

// ---------------- edge messages: acc[dst] += tr[r][src] / max(cnt,1) ----------------
// 16 lanes per edge, b128 row gather + f32 atomic scatter.
__global__ void msg_kernel(const int* __restrict__ src, const int* __restrict__ dst,
                           const int* __restrict__ et, const unsigned* __restrict__ cnt,
                           const float* __restrict__ tr, float* __restrict__ acc,
                           int E, int N) {
  long gid = (long)blockIdx.x * blockDim.x + threadIdx.x;
  long e = gid >> 4;
  int  q = (int)(gid & 15);
  if (e < E) {
    int s = src[e], t = dst[e], r = et[e];
    unsigned c = cnt[(long)t * NRELS + r];
    float inv = 1.0f / (float)(c ? c : 1u);
    const v4f* row = (const v4f*)(tr + ((long)r * N + s) * NDIM);
    v4f v = row[q];                                   // global_load_b128
    float* ap = acc + (long)t * NDIM + q * 4;
    unsafeAtomicAdd(ap + 0, v.x * inv);
    unsafeAtomicAdd(ap + 1, v.y * inv);
    unsafeAtomicAdd(ap + 2, v.z * inv);
    unsafeAtomicAdd(ap + 3, v.w * inv);
  }
}

__global__ void relu_kernel(v4f* __restrict__ p, int n4) {
  int i = blockIdx.x * blockDim.x + threadIdx.x;
  if (i < n4) {
    v4f v = p[i];
    v.x = fmaxf(v.x, 0.0f); v.y = fmaxf(v.y, 0.0f);
    v.z = fmaxf(v.z, 0.0f); v.w = fmaxf(v.w, 0.0f);
    p[i] = v;
  }
}

// ---------------- global mean pool ----------------
__global__ void pool_kernel(const float* __restrict__ h, const int* __restrict__ batch,
                            float* __restrict__ gsum, float* __restrict__ gcnt, int n16) {
  int i = blockIdx.x * blockDim.x + threadIdx.x;
  if (i < n16) {
    int n = i >> 4, q = i & 15;
    int g = batch[n];
    const v4f* row = (const v4f*)(h + (long)n * NDIM);
    v4f v = row[q];
    float* gp = gsum + (long)g * NDIM + q * 4;
    unsafeAtomicAdd(gp + 0, v.x);
    unsafeAtomicAdd(gp + 1, v.y);
    unsafeAtomicAdd(gp + 2, v.z);
    unsafeAtomicAdd(gp + 3, v.w);
    if (q == 0) unsafeAtomicAdd(&gcnt[g], 1.0f);
  }
}

// ---------------- head: out = (gsum/gcnt) @ lin_w + lin_b ----------------
__global__ void head_kernel(const float* __restrict__ gsum, const float* __restrict__ gcnt,
                            const float* __restrict__ lw, const float* __restrict__ lb,
                            float* __restrict__ out, int G) {
  int tid = blockIdx.x * blockDim.x + threadIdx.x;
  if (tid < G * 2) {
    int g = tid >> 1, c = tid & 1;
    float inv = 1.0f / fmaxf(gcnt[g], 1.0f);
    float s = 0.0f;
#pragma unroll
    for (int k = 0; k < NDIM; ++k) s += gsum[g * NDIM + k] * lw[k * 2 + c];
    out[tid] = s * inv + lb[c];
  }
}

extern "C" void kernel_launch(void* const* d_in, const int* in_sizes, int n_in,
                              void* d_out, int out_size, void* d_ws, size_t ws_size,
                              hipStream_t stream) {
  (void)n_in; (void)ws_size;
  const int*   x     = (const int*)d_in[0];
  const int*   ei    = (const int*)d_in[1];   // [2][E]: row0=src, row1=dst
  const int*   et    = (const int*)d_in[2];
  const int*   batch = (const int*)d_in[3];
  const float* ew    = (const float*)d_in[4];
  const float* W1    = (const float*)d_in[5];
  const float* root1 = (const float*)d_in[6];
  const float* b1    = (const float*)d_in[7];
  const float* W2    = (const float*)d_in[8];
  const float* root2 = (const float*)d_in[9];
  const float* b2    = (const float*)d_in[10];
  const float* lw    = (const float*)d_in[11];
  const float* lb    = (const float*)d_in[12];
  float* out = (float*)d_out;

  const int N = in_sizes[0];
  const int E = in_sizes[2];
  const int G = out_size / 2;
  const int n64 = N * NDIM;
  const int ntiles = (N + 15) / 16;

  // workspace layout
  float*    h    = (float*)d_ws;               // [N][64]  (ping)
  float*    tr   = h + (size_t)n64;            // [3][N][64]
  float*    acc  = tr + (size_t)3 * n64;       // [N][64]  (pong)
  unsigned* cnt  = (unsigned*)(acc + (size_t)n64);       // [N][3]
  float*    gsum = (float*)(cnt + (size_t)N * NRELS);    // [G][64]
  float*    gcnt = gsum + (size_t)G * NDIM;              // [G]

  const int* srcp = ei;
  const int* dstp = ei + E;

  const unsigned msgBlocks = (unsigned)(((long)E * 16 + 255) / 256);
  const unsigned gemmBlocks = (unsigned)((ntiles + 7) / 8);

  embed_kernel<<<(n64 + 255) / 256, 256, 0, stream>>>(x, ew, h, n64);

  // per-(dst,rel) in-degrees (identical for both layers)
  zero_u32_kernel<<<(N * NRELS + 255) / 256, 256, 0, stream>>>(cnt, N * NRELS);
  count_kernel<<<(E + 255) / 256, 256, 0, stream>>>(dstp, et, cnt, E);

  // ---- layer 1: h -> acc ----
  rgcn_gemm_kernel<<<gemmBlocks, 256, 0, stream>>>(h, W1, root1, b1, tr, acc, N, ntiles);
  msg_kernel<<<msgBlocks, 256, 0, stream>>>(srcp, dstp, et, cnt, tr, acc, E, N);
  relu_kernel<<<(n64 / 4 + 255) / 256, 256, 0, stream>>>((v4f*)acc, n64 / 4);

  // ---- layer 2: acc -> h ----
  rgcn_gemm_kernel<<<gemmBlocks, 256, 0, stream>>>(acc, W2, root2, b2, tr, h, N, ntiles);
  msg_kernel<<<msgBlocks, 256, 0, stream>>>(srcp, dstp, et, cnt, tr, h, E, N);
  relu_kernel<<<(n64 / 4 + 255) / 256, 256, 0, stream>>>((v4f*)h, n64 / 4);

  // ---- pool + head ----
  zero_f32_kernel<<<(G * NDIM + G + 255) / 256, 256, 0, stream>>>(gsum, G * NDIM + G);
  pool_kernel<<<(N * 16 + 255) / 256, 256, 0, stream>>>(h, batch, gsum, gcnt, N * 16);
  head_kernel<<<(G * 2 + 255) / 256, 256, 0, stream>>>(gsum, gcnt, lw, lb, out, G);
}